// GraphAttention_35682588295310
// MI455X (gfx1250) — compile-verified
//
#include <hip/hip_runtime.h>
#include <hip/hip_bf16.h>
#include <math.h>

// ---------------------------------------------------------------------------
// GAT layer: h = x@W ; alpha = LReLU(s_i[dst]+s_j[src]) ; softmax over dst ;
// out[dst] += h[src]*coef ; + bias
// Sizes fixed by the reference: N=50000, E=800000, IN=128, H=4, O=32.
// ---------------------------------------------------------------------------

#define N_NODES   50000
#define E_EDGES   800000
#define IN_F      128
#define H_HEADS   4
#define O_F       32
#define OUT_F     128          // H*O
#define NEG_SLOPE 0.2f

typedef float v2f __attribute__((ext_vector_type(2)));
typedef float v8f __attribute__((ext_vector_type(8)));

// ---------------------------------------------------------------------------
// Kernel 0: init amax=-inf, denom=0, out = broadcast(bias)
// ---------------------------------------------------------------------------
__global__ __launch_bounds__(256) void gat_init(float* __restrict__ amax,
                                                float* __restrict__ denom,
                                                float* __restrict__ out,
                                                const float* __restrict__ bias) {
    const int idx = blockIdx.x * 256 + threadIdx.x;
    const int total = N_NODES * OUT_F;            // 6.4M
    if (idx < total) {
        out[idx] = bias[idx & (OUT_F - 1)];
    }
    if (idx < N_NODES * H_HEADS) {                // 200K
        amax[idx]  = -__builtin_inff();
        denom[idx] = 0.0f;
    }
}

// ---------------------------------------------------------------------------
// Kernel 1: h = x @ W   (50000x128 @ 128x128, f32, full-precision WMMA)
// One wave computes one 16x16 output tile via 32x V_WMMA_F32_16X16X4_F32.
// Block = 256 threads = 8 waves = one 16-row slab covering all 128 columns.
// ---------------------------------------------------------------------------
__global__ __launch_bounds__(256) void gat_gemm(const float* __restrict__ x,
                                                const float* __restrict__ w,
                                                float* __restrict__ h) {
    __shared__ float As[16 * 132];                // padded stride: 132%64=4 -> no bank conflicts
    const int m0  = blockIdx.x * 16;
    const int tid = threadIdx.x;

    // Cooperative, coalesced stage of the 16x128 A slab.
    #pragma unroll
    for (int i = tid; i < 16 * 128; i += 256) {
        const int r = i >> 7, c = i & 127;
        As[r * 132 + c] = x[(m0 + r) * IN_F + c];
    }
    __syncthreads();

    const int wave = tid >> 5;
    const int lane = tid & 31;
    const int n0   = wave * 16;                   // column tile
    const int half = lane >> 4;                   // selects K pair {0,1} vs {2,3}
    const int l16  = lane & 15;

    v8f acc = {0.f, 0.f, 0.f, 0.f, 0.f, 0.f, 0.f, 0.f};

    #pragma unroll
    for (int k0 = 0; k0 < IN_F; k0 += 4) {
        // A 16x4 fragment: lane l -> M=l%16 ; comp c -> K = k0 + 2*half + c
        v2f a, b;
        a.x = As[l16 * 132 + k0 + 2 * half + 0];
        a.y = As[l16 * 132 + k0 + 2 * half + 1];
        // B 4x16 fragment (row striped across lanes): comp c -> K = k0 + 2*half + c, N = l16
        b.x = w[(k0 + 2 * half + 0) * OUT_F + n0 + l16];
        b.y = w[(k0 + 2 * half + 1) * OUT_F + n0 + l16];
        acc = __builtin_amdgcn_wmma_f32_16x16x4_f32(
            /*neg_a=*/false, a, /*neg_b=*/false, b,
            /*c_mod=*/(short)0, acc, /*reuse_a=*/false, /*reuse_b=*/false);
    }

    // C/D layout: VGPR r -> M = r + 8*half, N = l16
    const int col = n0 + l16;
    #pragma unroll
    for (int r = 0; r < 8; ++r) {
        h[(m0 + r + 8 * half) * OUT_F + col] = acc[r];
    }
}

// ---------------------------------------------------------------------------
// Kernel 2: per-node attention dots  s_i[n,hd] = <h[n,hd,:], a_i[hd,:]>,
//                                    s_j[n,hd] = <h[n,hd,:], a_j[hd,:]>
// (factors the reference's per-edge einsums into O(N*H*O) work)
// ---------------------------------------------------------------------------
__global__ __launch_bounds__(256) void gat_node_scores(const float* __restrict__ h,
                                                       const float* __restrict__ att,
                                                       float* __restrict__ s_i,
                                                       float* __restrict__ s_j) {
    const int idx = blockIdx.x * 256 + threadIdx.x;
    if (idx >= N_NODES * H_HEADS) return;
    const int n  = idx >> 2;
    const int hd = idx & 3;
    const float* hp = h + n * OUT_F + hd * O_F;
    const float* ai = att + hd * (2 * O_F);
    const float* aj = ai + O_F;
    float si = 0.f, sj = 0.f;
    #pragma unroll
    for (int o = 0; o < O_F; ++o) {
        const float hv = hp[o];
        si = fmaf(hv, ai[o], si);
        sj = fmaf(hv, aj[o], sj);
    }
    s_i[idx] = si;
    s_j[idx] = sj;
}

// Mixed sign-split float atomic-max via int punning (init must be -inf).
__device__ __forceinline__ void atomicMaxF32(float* addr, float val) {
    if (val >= 0.f) atomicMax((int*)addr, __float_as_int(val));
    else            atomicMin((unsigned int*)addr, __float_as_uint(val));
}

// ---------------------------------------------------------------------------
// Kernel 3: alpha = leaky_relu(s_i[dst]+s_j[src]); segment max over dst
// ---------------------------------------------------------------------------
__global__ __launch_bounds__(256) void gat_edge_alpha(const long long* __restrict__ ei,
                                                      const float* __restrict__ s_i,
                                                      const float* __restrict__ s_j,
                                                      float* __restrict__ alpha,
                                                      float* __restrict__ amax) {
    const int e = blockIdx.x * 256 + threadIdx.x;
    if (e >= E_EDGES) return;
    const int src = (int)ei[e];
    const int dst = (int)ei[E_EDGES + e];
    #pragma unroll
    for (int hd = 0; hd < H_HEADS; ++hd) {
        float a = s_i[dst * H_HEADS + hd] + s_j[src * H_HEADS + hd];
        a = (a > 0.f) ? a : NEG_SLOPE * a;
        alpha[e * H_HEADS + hd] = a;
        atomicMaxF32(&amax[dst * H_HEADS + hd], a);
    }
}

// ---------------------------------------------------------------------------
// Kernel 4: ex = exp(alpha - amax[dst]); segment sum -> denom  (ex overwrites alpha)
// ---------------------------------------------------------------------------
__global__ __launch_bounds__(256) void gat_edge_exp(const long long* __restrict__ ei,
                                                    float* __restrict__ alpha,
                                                    const float* __restrict__ amax,
                                                    float* __restrict__ denom) {
    const int e = blockIdx.x * 256 + threadIdx.x;
    if (e >= E_EDGES) return;
    const int dst = (int)ei[E_EDGES + e];
    #pragma unroll
    for (int hd = 0; hd < H_HEADS; ++hd) {
        const float ex = __expf(alpha[e * H_HEADS + hd] - amax[dst * H_HEADS + hd]);
        alpha[e * H_HEADS + hd] = ex;
        atomicAdd(&denom[dst * H_HEADS + hd], ex);
    }
}

// ---------------------------------------------------------------------------
// Kernel 5: out[dst,:] += h[src,:] * coef   (2 edges per 256-thread block;
// h fits in L2 so the src gathers stay on-chip)
// ---------------------------------------------------------------------------
__global__ __launch_bounds__(256) void gat_edge_aggregate(const long long* __restrict__ ei,
                                                          const float* __restrict__ h,
                                                          const float* __restrict__ ex,
                                                          const float* __restrict__ denom,
                                                          float* __restrict__ out) {
    const int t = threadIdx.x;
    const int e = blockIdx.x * 2 + (t >> 7);
    const int c = t & 127;
    const int hd = c >> 5;
    const int src = (int)ei[e];
    const int dst = (int)ei[E_EDGES + e];
    const float coef = ex[e * H_HEADS + hd] / (denom[dst * H_HEADS + hd] + 1e-16f);
    atomicAdd(&out[dst * OUT_F + c], h[src * OUT_F + c] * coef);
}

// ---------------------------------------------------------------------------
extern "C" void kernel_launch(void* const* d_in, const int* in_sizes, int n_in,
                              void* d_out, int out_size, void* d_ws, size_t ws_size,
                              hipStream_t stream) {
    const float*     x    = (const float*)d_in[0];
    const long long* ei   = (const long long*)d_in[1];   // int64 (2,E)
    const float*     w    = (const float*)d_in[2];       // (128,4,32) == (128,128) row-major
    const float*     att  = (const float*)d_in[3];       // (4,64)
    const float*     bias = (const float*)d_in[4];       // (128,)
    float*           out  = (float*)d_out;

    // Workspace layout (256B-aligned offsets)
    char* ws = (char*)d_ws;
    float* h     = (float*)(ws);                                  // 25,600,000 B
    float* s_i   = (float*)(ws + 25600000);                       //    800,000 B
    float* s_j   = (float*)(ws + 26400000);                       //    800,000 B
    float* amax  = (float*)(ws + 27200000);                       //    800,000 B
    float* denom = (float*)(ws + 28000000);                       //    800,000 B
    float* alpha = (float*)(ws + 28800000);                       // 12,800,000 B
    (void)in_sizes; (void)n_in; (void)out_size; (void)ws_size;

    // 0) init out=bias, amax=-inf, denom=0 (re-done every call: deterministic)
    gat_init<<<(N_NODES * OUT_F + 255) / 256, 256, 0, stream>>>(amax, denom, out, bias);

    // 1) h = x @ W   (3125 slabs of 16 rows; N is an exact multiple of 16)
    gat_gemm<<<N_NODES / 16, 256, 0, stream>>>(x, w, h);

    // 2) per-node attention scores
    gat_node_scores<<<(N_NODES * H_HEADS + 255) / 256, 256, 0, stream>>>(h, att, s_i, s_j);

    // 3) alpha + segment max      (E is an exact multiple of 256)
    gat_edge_alpha<<<E_EDGES / 256, 256, 0, stream>>>(ei, s_i, s_j, alpha, amax);

    // 4) exp + segment sum
    gat_edge_exp<<<E_EDGES / 256, 256, 0, stream>>>(ei, alpha, amax, denom);

    // 5) weighted aggregation into out (bias already folded in)
    gat_edge_aggregate<<<E_EDGES / 2, 256, 0, stream>>>(ei, h, alpha, denom, out);
}